// SDPAttention_60215441490285
// MI455X (gfx1250) — compile-verified
//
#include <hip/hip_runtime.h>
#include <hip/hip_bf16.h>

typedef float v2f __attribute__((ext_vector_type(2)));
typedef float v8f __attribute__((ext_vector_type(8)));

#define B_  4
#define H_  16
#define S_  1024
#define DK_ 64
#define SCALE_ 0.125f   // 1/sqrt(64)

__global__ __launch_bounds__(128) void sdpa_wmma_f32_kernel(
    const float* __restrict__ Q, const float* __restrict__ K,
    const float* __restrict__ V, const int* __restrict__ Mask,
    float* __restrict__ Out, float* __restrict__ Attn)
{
    // Per-wave 16x16 score tile (re-layout C->A between the two matmuls),
    // plus per-row softmax stats for the finalize pass.
    __shared__ __align__(16) float tile[4][16][16];
    __shared__ float mA[64];
    __shared__ float invA[64];

    const int tid  = threadIdx.x;
    const int wave = tid >> 5;
    const int lane = tid & 31;
    const int l16  = lane & 15;
    const int hi   = lane >> 4;      // 0: lanes 0-15, 1: lanes 16-31

    const int qblk = blockIdx.x & 15;     // S/64 = 16 q-blocks
    const int bh   = blockIdx.x >> 4;     // b*H + h
    const int b    = bh >> 4;             // H = 16

    const float* Qp = Q + (size_t)bh * S_ * DK_;
    const float* Kp = K + (size_t)bh * S_ * DK_;
    const float* Vp = V + (size_t)bh * S_ * DK_;
    const int*   Mp = Mask + (size_t)b * S_;
    float* Outp  = Out  + (size_t)bh * S_ * DK_;
    float* Attnp = Attn + (size_t)bh * S_ * S_;

    const int qrow0 = qblk * 64 + wave * 16;   // this wave's 16 query rows

    // ---- Preload Q A-fragments for all 16 reduction steps (DK=64, K=4/step).
    // A layout (16x4 f32): lanes 0-15 hold K=0,1 ; lanes 16-31 hold K=2,3.
    v2f aQ[16];
    {
        const size_t ra = (size_t)(qrow0 + l16) * DK_;
        #pragma unroll
        for (int t = 0; t < 16; ++t) {
            const float2 qv = *(const float2*)(Qp + ra + 4 * t + 2 * hi);
            aQ[t].x = qv.x; aQ[t].y = qv.y;
        }
    }

    const v8f zero8 = {0.f, 0.f, 0.f, 0.f, 0.f, 0.f, 0.f, 0.f};
    v8f o[4];                 // O accumulators: 4 x (16x16) covering DK=64
    #pragma unroll
    for (int nb = 0; nb < 4; ++nb) o[nb] = zero8;

    float m_run = -INFINITY;  // running row max (lanes 0-15, one row each)
    float s_run = 0.0f;       // running sum of exp(x - m_run)

    for (int kb = 0; kb < 64; ++kb) {
        // ---- score tile = Q(16xDK) . K^T(DKx16) via 16 f32 WMMAs ----
        v8f c = zero8;
        {
            const size_t kr = (size_t)(kb * 16 + l16) * DK_;
            #pragma unroll
            for (int t = 0; t < 16; ++t) {
                const float2 kv = *(const float2*)(Kp + kr + 4 * t + 2 * hi);
                v2f bK; bK.x = kv.x; bK.y = kv.y;
                c = __builtin_amdgcn_wmma_f32_16x16x4_f32(
                        false, aQ[t], false, bK, (short)0, c, false, false);
            }
        }
        if (kb + 1 < 64)  // gfx1250 global_prefetch_b8 on next K tile
            __builtin_prefetch(Kp + (size_t)((kb + 1) * 16 + l16) * DK_, 0, 1);

        // ---- scale + mask (each lane owns column n = kb*16 + l16) ----
        const int  col  = kb * 16 + l16;
        const bool keep = (Mp[col] != 0);
        float sc[8];
        #pragma unroll
        for (int i = 0; i < 8; ++i) {
            const float x = c[i] * SCALE_;
            sc[i] = keep ? x : -10000.0f;
        }

        __syncthreads();  // WAR on the LDS tile vs previous iteration

        // C/D layout: VGPR i -> row (i + 8*hi), col l16.
        // Stage tile in LDS and also emit raw masked scores (scratch that
        // the softmax finalize pass overwrites with normalized weights).
        #pragma unroll
        for (int i = 0; i < 8; ++i) {
            const int r = i + 8 * hi;
            tile[wave][r][l16] = sc[i];
            Attnp[(size_t)(qrow0 + r) * S_ + col] = sc[i];
        }

        __syncthreads();  // tile visible

        // ---- online softmax stats: lanes 0-15 each own one row ----
        if (lane < 16) {
            float tmax = -INFINITY;
            #pragma unroll
            for (int j = 0; j < 16; ++j)
                tmax = fmaxf(tmax, tile[wave][lane][j]);
            const float mnew = fmaxf(m_run, tmax);
            float acc = 0.0f;
            #pragma unroll
            for (int j = 0; j < 16; ++j)
                acc += __expf(tile[wave][lane][j] - mnew);
            s_run = s_run * __expf(m_run - mnew) + acc;
            m_run = mnew;
        }

        // ---- O += S_tile(16x16) . V_tile(16x64): 16 f32 WMMAs ----
        #pragma unroll
        for (int kk = 0; kk < 4; ++kk) {
            const float2 av = *(const float2*)&tile[wave][l16][4 * kk + 2 * hi];
            v2f aS; aS.x = av.x; aS.y = av.y;
            const int vr = kb * 16 + 4 * kk + 2 * hi;
            #pragma unroll
            for (int nb = 0; nb < 4; ++nb) {
                v2f bV;
                bV.x = Vp[(size_t)vr * DK_ + nb * 16 + l16];
                bV.y = Vp[(size_t)(vr + 1) * DK_ + nb * 16 + l16];
                o[nb] = __builtin_amdgcn_wmma_f32_16x16x4_f32(
                            false, aS, false, bV, (short)0, o[nb], false, false);
            }
        }
    }

    // ---- write O = qk . V (faithful to reference's raw-score bug) ----
    #pragma unroll
    for (int nb = 0; nb < 4; ++nb) {
        #pragma unroll
        for (int i = 0; i < 8; ++i) {
            const int r = i + 8 * hi;
            Outp[(size_t)(qrow0 + r) * DK_ + nb * 16 + l16] = o[nb][i];
        }
    }

    // ---- softmax finalize: publish per-row (m, 1/sum) then normalize ----
    if (lane < 16) {
        mA[wave * 16 + lane]   = m_run;
        invA[wave * 16 + lane] = 1.0f / s_run;
    }
    __syncthreads();  // workgroup fence: LDS stats + our global raw stores

    // 16 rows x 1024 cols per wave, coalesced 32-lane chunks.
    for (int it = 0; it < 512; ++it) {
        const int idx  = it * 32 + lane;
        const int row  = idx >> 10;
        const int ccol = idx & 1023;
        float* p = Attnp + (size_t)(qrow0 + row) * S_ + ccol;
        const float x = *p;
        *p = __expf(x - mA[wave * 16 + row]) * invA[wave * 16 + row];
    }
}

extern "C" void kernel_launch(void* const* d_in, const int* in_sizes, int n_in,
                              void* d_out, int out_size, void* d_ws, size_t ws_size,
                              hipStream_t stream) {
    (void)in_sizes; (void)n_in; (void)out_size; (void)d_ws; (void)ws_size;
    const float* q    = (const float*)d_in[0];
    const float* k    = (const float*)d_in[1];
    const float* v    = (const float*)d_in[2];
    const int*   mask = (const int*)d_in[3];

    float* out  = (float*)d_out;                               // [B,H,S,DK]
    float* attn = out + (size_t)B_ * H_ * S_ * DK_;            // [B,H,S,S]

    const dim3 grid(B_ * H_ * (S_ / 64));   // 1024 workgroups
    const dim3 block(128);                  // 4 wave32s
    sdpa_wmma_f32_kernel<<<grid, block, 0, stream>>>(q, k, v, mask, out, attn);
}